// Attention_9844065042495
// MI455X (gfx1250) — compile-verified
//
#include <hip/hip_runtime.h>

// ---------------------------------------------------------------------------
// Transformer-XL attention with expire-span, fused flash-style for gfx1250.
// fp32 WMMA (V_WMMA_F32_16X16X4_F32) everywhere; attention stages K/V tiles
// in LDS with double-buffered GLOBAL_LOAD_ASYNC_TO_LDS_B128 (ASYNCcnt).
// ---------------------------------------------------------------------------

typedef __attribute__((ext_vector_type(2))) float v2f;
typedef __attribute__((ext_vector_type(8))) float v8f;

#define WMMA4(a, b, c) \
  __builtin_amdgcn_wmma_f32_16x16x4_f32(false, (a), false, (b), (short)0, (c), false, false)

constexpr int BB   = 2;
constexpr int NN   = 2048;   // query length
constexpr int DIM  = 1024;
constexpr int HH   = 16;
constexpr int DH   = 64;
constexpr int MEML = 1024;
constexpr int SS   = MEML + NN;       // 3072 keys
constexpr float SCALE = 0.125f;       // dh^-0.5

__device__ __forceinline__ v2f ld2(const float* p) { return *(const v2f*)p; }

// async copy of 16 bytes global -> LDS (GVS mode: SGPR64 base + 32-bit offset)
__device__ __forceinline__ void async_ld128(unsigned ldsOff, unsigned gOff,
                                            const float* base) {
  asm volatile("global_load_async_to_lds_b128 %0, %1, %2"
               :: "v"(ldsOff), "v"(gOff), "s"(base) : "memory");
}
__device__ __forceinline__ void wait_async0() {
  asm volatile("s_wait_asynccnt 0x0" ::: "memory");
}
__device__ __forceinline__ unsigned lds_off(const void* p) {
  return (unsigned)(uintptr_t)p;   // LDS aperture: low 32 bits = LDS byte offset
}

// Fragment conventions (wave32):
//  A 16x4 (MxK): lane L holds rows M=L%16; VGPR0 = K { +0 | +2 }, VGPR1 = K { +1 | +3 }
//  B 4x16 (KxN): lane L holds col N=L%16; VGPR0 = K { +0 | +2 }, VGPR1 = K { +1 | +3 }
//  C 16x16:      vgpr v, lane L -> (row = v + 8*(L/16), col = L%16)

// ------------------------- q = (x @ Wq + bq) * scale -----------------------
__global__ __launch_bounds__(128) void qproj_k(const float* __restrict__ x,
                                               const float* __restrict__ Wq,
                                               const float* __restrict__ bq,
                                               float* __restrict__ q) {
  const int lane = threadIdx.x & 31, lm = lane & 15, lh = lane >> 4;
  const int gw = (blockIdx.x * blockDim.x + threadIdx.x) >> 5;
  const int NT = DIM / 16;                       // 64
  if (gw >= (BB * NN / 16) * NT) return;
  const int m0 = (gw / NT) * 16, n0 = (gw % NT) * 16;

  const float* arow = x + (size_t)(m0 + lm) * DIM + 2 * lh;
  const float* bcol = Wq + (size_t)(2 * lh) * DIM + n0 + lm;
  v8f c = {0.f, 0.f, 0.f, 0.f, 0.f, 0.f, 0.f, 0.f};
#pragma unroll 4
  for (int k0 = 0; k0 < DIM; k0 += 4) {
    if ((k0 & 15) == 0) __builtin_prefetch(arow + k0 + 64, 0, 0);
    v2f a = ld2(arow + k0);
    v2f b = {bcol[(size_t)k0 * DIM], bcol[(size_t)(k0 + 1) * DIM]};
    c = WMMA4(a, b, c);
  }
  const int col = n0 + lm, h = col >> 6, d = col & 63;
  const int bi = m0 / NN, ibase = m0 % NN;
  const float bias = bq[col];
#pragma unroll
  for (int vv = 0; vv < 8; ++vv) {
    int i = ibase + vv + 8 * lh;
    q[((((size_t)bi * HH + h) * NN) + i) * DH + d] = (c[vv] + bias) * SCALE;
  }
}

// --------------------- k,v = concat(mem,x) @ Wkv + bkv ---------------------
__global__ __launch_bounds__(128) void kvproj_k(const float* __restrict__ x,
                                                const float* __restrict__ mem,
                                                const float* __restrict__ Wkv,
                                                const float* __restrict__ bkv,
                                                float* __restrict__ kbuf,
                                                float* __restrict__ vbuf) {
  const int lane = threadIdx.x & 31, lm = lane & 15, lh = lane >> 4;
  const int gw = (blockIdx.x * blockDim.x + threadIdx.x) >> 5;
  const int NT = 2 * DIM / 16;                   // 128
  if (gw >= (BB * SS / 16) * NT) return;
  const int m0 = (gw / NT) * 16, n0 = (gw % NT) * 16;

  const int bi = m0 / SS, tbase = m0 % SS;
  const int t = tbase + lm;
  const float* arow = (t < MEML) ? (mem + ((size_t)bi * MEML + t) * DIM)
                                 : (x + ((size_t)bi * NN + (t - MEML)) * DIM);
  arow += 2 * lh;
  const float* bcol = Wkv + (size_t)(2 * lh) * (2 * DIM) + n0 + lm;
  v8f c = {0.f, 0.f, 0.f, 0.f, 0.f, 0.f, 0.f, 0.f};
#pragma unroll 4
  for (int k0 = 0; k0 < DIM; k0 += 4) {
    if ((k0 & 15) == 0) __builtin_prefetch(arow + k0 + 64, 0, 0);
    v2f a = ld2(arow + k0);
    v2f b = {bcol[(size_t)k0 * (2 * DIM)], bcol[(size_t)(k0 + 1) * (2 * DIM)]};
    c = WMMA4(a, b, c);
  }
  const int col = n0 + lm;
  const float bias = bkv[col];
  float* dest = (col < DIM) ? kbuf : vbuf;
  const int cc = (col < DIM) ? col : (col - DIM);
  const int h = cc >> 6, d = cc & 63;
#pragma unroll
  for (int vv = 0; vv < 8; ++vv) {
    int tr = tbase + vv + 8 * lh;
    dest[((((size_t)bi * HH + h) * SS) + tr) * DH + d] = c[vv] + bias;
  }
}

// --------------------------- pos = pos_emb @ Wp + bp -----------------------
__global__ __launch_bounds__(128) void pproj_k(const float* __restrict__ pe,
                                               const float* __restrict__ Wp,
                                               const float* __restrict__ bp,
                                               float* __restrict__ pos) {
  const int lane = threadIdx.x & 31, lm = lane & 15, lh = lane >> 4;
  const int gw = (blockIdx.x * blockDim.x + threadIdx.x) >> 5;
  const int NT = DH / 16;                        // 4
  if (gw >= (NN / 16) * NT) return;
  const int m0 = (gw / NT) * 16, n0 = (gw % NT) * 16;

  const float* arow = pe + (size_t)(m0 + lm) * DIM + 2 * lh;
  const float* bcol = Wp + (size_t)(2 * lh) * DH + n0 + lm;
  v8f c = {0.f, 0.f, 0.f, 0.f, 0.f, 0.f, 0.f, 0.f};
#pragma unroll 4
  for (int k0 = 0; k0 < DIM; k0 += 4) {
    if ((k0 & 15) == 0) __builtin_prefetch(arow + k0 + 64, 0, 0);
    v2f a = ld2(arow + k0);
    v2f b = {bcol[(size_t)k0 * DH], bcol[(size_t)(k0 + 1) * DH]};
    c = WMMA4(a, b, c);
  }
  const int col = n0 + lm;
  const float bias = bp[col];
#pragma unroll
  for (int vv = 0; vv < 8; ++vv)
    pos[(size_t)(m0 + vv + 8 * lh) * DH + col] = c[vv] + bias;
}

// ------------- fused flash attention with XL rel-pos + expire mask ---------
// Workgroup = 8 waves = 8 consecutive query tiles of one (b,h).
// K/V tiles double-buffered in LDS via async global->LDS loads (ASYNCcnt).
__global__ __launch_bounds__(256) void attn_k(const float* __restrict__ qb,
                                              const float* __restrict__ kb,
                                              const float* __restrict__ vb,
                                              const float* __restrict__ pos,
                                              const float* __restrict__ em,
                                              float* __restrict__ ob) {
  __shared__ __align__(16) float kT[2][16][64];   // 8 KB
  __shared__ __align__(16) float vT[2][16][64];   // 8 KB
  __shared__ float scr[8][16][36];                // per-wave T-band / P relayout

  const int lane = threadIdx.x & 31, lm = lane & 15, lh = lane >> 4;
  const int w = threadIdx.x >> 5;
  const int bh = blockIdx.x >> 4;                 // (b*H+h)
  const int qt_base = (blockIdx.x & 15) * 8;
  const int qt = qt_base + w;
  const int bi = bh >> 4;                         // / HH
  const int i0 = qt * 16;

  const float* Q = qb + ((size_t)bh * NN + i0) * DH;
  const float* K = kb + (size_t)bh * SS * DH;
  const float* V = vb + (size_t)bh * SS * DH;

  // staging map: 256 threads x 16B cover one 16x64 fp32 tile
  const int srow = threadIdx.x >> 4;              // 0..15
  const int scol = (threadIdx.x & 15) * 4;        // 0..60

  // Q tile as 16 A-fragments (K-steps of 4 over dh=64), kept in registers
  v2f aq[16];
#pragma unroll
  for (int kk = 0; kk < 16; ++kk)
    aq[kk] = ld2(Q + lm * DH + kk * 4 + 2 * lh);

  float mrow[8], lrow[8];
  v8f acc[4];
#pragma unroll
  for (int vv = 0; vv < 8; ++vv) { mrow[vv] = -3.0e38f; lrow[vv] = 0.f; }
#pragma unroll
  for (int dt = 0; dt < 4; ++dt)
    acc[dt] = (v8f){0.f, 0.f, 0.f, 0.f, 0.f, 0.f, 0.f, 0.f};

  const int jt_wlast = MEML / 16 + qt;            // this wave's last causal tile
  const int jt_glast = MEML / 16 + qt_base + 7;   // group max

  // prologue: async-stage tile 0 into buffer 0
  {
    unsigned go = (unsigned)(((0 + srow) * DH + scol) * 4);
    async_ld128(lds_off(&kT[0][srow][scol]), go, K);
    async_ld128(lds_off(&vT[0][srow][scol]), go, V);
  }

  for (int jt = 0; jt <= jt_glast; ++jt) {
    const int buf = jt & 1;
    const int j0 = jt * 16;
    wait_async0();
    __syncthreads();                              // tile jt resident for all waves
    if (jt < jt_glast) {                          // overlap: fetch tile jt+1
      unsigned go = (unsigned)((((jt + 1) * 16 + srow) * DH + scol) * 4);
      async_ld128(lds_off(&kT[buf ^ 1][srow][scol]), go, K);
      async_ld128(lds_off(&vT[buf ^ 1][srow][scol]), go, V);
    }

    if (jt <= jt_wlast) {                         // wave-uniform causal guard
      // ---- S = Q K^T  (q pre-scaled by dh^-0.5), K from LDS
      v8f s = {0.f, 0.f, 0.f, 0.f, 0.f, 0.f, 0.f, 0.f};
#pragma unroll
      for (int kk = 0; kk < 16; ++kk)
        s = WMMA4(aq[kk], *(const v2f*)&kT[buf][lm][kk * 4 + 2 * lh], s);

      // ---- relative-position term, x-region key tiles only
      if (j0 >= MEML) {
        const int base = (j0 - MEML) - i0 + (NN - 1);
        int pr0 = base - 15 + lm;       if (pr0 > NN - 1) pr0 = NN - 1;
        int pr1 = base - 15 + 16 + lm;  if (pr1 > NN - 1) pr1 = NN - 1;
        const float* p0 = pos + (size_t)pr0 * DH + 2 * lh;
        const float* p1 = pos + (size_t)pr1 * DH + 2 * lh;
        v8f t0 = {0.f, 0.f, 0.f, 0.f, 0.f, 0.f, 0.f, 0.f};
        v8f t1 = {0.f, 0.f, 0.f, 0.f, 0.f, 0.f, 0.f, 0.f};
#pragma unroll
        for (int kk = 0; kk < 16; ++kk) {
          t0 = WMMA4(aq[kk], ld2(p0 + kk * 4), t0);
          t1 = WMMA4(aq[kk], ld2(p1 + kk * 4), t1);
        }
        // stage T in per-wave LDS, gather diagonal R[ii,jj] = T[ii, jj-ii+15]
#pragma unroll
        for (int vv = 0; vv < 8; ++vv) {
          scr[w][vv + 8 * lh][lm]      = t0[vv];
          scr[w][vv + 8 * lh][16 + lm] = t1[vv];
        }
        asm volatile("s_wait_dscnt 0" ::: "memory");
#pragma unroll
        for (int vv = 0; vv < 8; ++vv) {
          int row = vv + 8 * lh;
          s[vv] += scr[w][row][lm - row + 15];
        }
        if (jt == jt_wlast) {  // diagonal tile: mask j > i
#pragma unroll
          for (int vv = 0; vv < 8; ++vv)
            if (lm > vv + 8 * lh) s[vv] = -3.0e38f;
        }
      }

      // expire-mask value for this lane's key column (mem region only)
      const float emv = (j0 < MEML) ? em[bi * MEML + j0 + lm] : 1.0f;

      // ---- online softmax (rows on VGPR index; reduce across 16-lane half)
      float tmax[8];
#pragma unroll
      for (int vv = 0; vv < 8; ++vv) tmax[vv] = s[vv];
#pragma unroll
      for (int off = 1; off < 16; off <<= 1)
#pragma unroll
        for (int vv = 0; vv < 8; ++vv)
          tmax[vv] = fmaxf(tmax[vv], __shfl_xor(tmax[vv], off, 32));
#pragma unroll
      for (int vv = 0; vv < 8; ++vv) {
        float mnew = fmaxf(mrow[vv], tmax[vv]);
        float corr = __expf(mrow[vv] - mnew);
        mrow[vv] = mnew;
        lrow[vv] *= corr;
#pragma unroll
        for (int dt = 0; dt < 4; ++dt) acc[dt][vv] *= corr;
        float p = __expf(s[vv] - mnew);
        lrow[vv] += p;                 // denominator excludes expire mask
        s[vv] = p * emv;               // numerator includes it
      }

      // ---- P@V: relayout P (C-layout -> A-layout) through per-wave LDS
#pragma unroll
      for (int vv = 0; vv < 8; ++vv) scr[w][vv + 8 * lh][lm] = s[vv];
      asm volatile("s_wait_dscnt 0" ::: "memory");
#pragma unroll
      for (int k0 = 0; k0 < 16; k0 += 4) {
        v2f ap = *(const v2f*)&scr[w][lm][k0 + 2 * lh];
        const int r0 = k0 + 2 * lh;
#pragma unroll
        for (int dt = 0; dt < 4; ++dt) {
          v2f b = {vT[buf][r0][dt * 16 + lm], vT[buf][r0 + 1][dt * 16 + lm]};
          acc[dt] = WMMA4(ap, b, acc[dt]);
        }
      }
    }
    __syncthreads();                   // everyone done before buf reused
  }

  // ---- finalize: row-sum of l across half, divide, store [b,h,n,dh]
#pragma unroll
  for (int off = 1; off < 16; off <<= 1)
#pragma unroll
    for (int vv = 0; vv < 8; ++vv)
      lrow[vv] += __shfl_xor(lrow[vv], off, 32);
#pragma unroll
  for (int vv = 0; vv < 8; ++vv) {
    float inv = 1.0f / lrow[vv];
    int row = i0 + vv + 8 * lh;
    float* orow = ob + ((size_t)bh * NN + row) * DH + lm;
#pragma unroll
    for (int dt = 0; dt < 4; ++dt) orow[dt * 16] = acc[dt][vv] * inv;
  }
}

// ------------------------- out = O @ Wo + bo  (gathered A) -----------------
__global__ __launch_bounds__(128) void oproj_k(const float* __restrict__ o,
                                               const float* __restrict__ Wo,
                                               const float* __restrict__ bo,
                                               float* __restrict__ out) {
  const int lane = threadIdx.x & 31, lm = lane & 15, lh = lane >> 4;
  const int gw = (blockIdx.x * blockDim.x + threadIdx.x) >> 5;
  const int NT = DIM / 16;
  if (gw >= (BB * NN / 16) * NT) return;
  const int m0 = (gw / NT) * 16, n0 = (gw % NT) * 16;

  const int bi = m0 / NN, i = (m0 % NN) + lm;
  const float* obase = o + (size_t)bi * HH * NN * DH + (size_t)i * DH;
  const float* bcol = Wo + (size_t)(2 * lh) * DIM + n0 + lm;
  v8f c = {0.f, 0.f, 0.f, 0.f, 0.f, 0.f, 0.f, 0.f};
#pragma unroll 4
  for (int k0 = 0; k0 < DIM; k0 += 4) {
    int ka = k0 + 2 * lh;
    const float* ap = obase + (size_t)(ka >> 6) * NN * DH + (ka & 63);
    if ((k0 & 15) == 0) __builtin_prefetch(ap + 16, 0, 0);
    v2f a = ld2(ap);
    v2f b = {bcol[(size_t)k0 * DIM], bcol[(size_t)(k0 + 1) * DIM]};
    c = WMMA4(a, b, c);
  }
  const int col = n0 + lm;
  const float bias = bo[col];
#pragma unroll
  for (int vv = 0; vv < 8; ++vv)
    out[(size_t)(m0 + vv + 8 * lh) * DIM + col] = c[vv] + bias;
}

// ---------------------------------------------------------------------------
extern "C" void kernel_launch(void* const* d_in, const int* in_sizes, int n_in,
                              void* d_out, int out_size, void* d_ws, size_t ws_size,
                              hipStream_t stream) {
  const float* x   = (const float*)d_in[0];
  const float* pe  = (const float*)d_in[1];
  const float* mem = (const float*)d_in[2];
  const float* em  = (const float*)d_in[3];
  const float* Wq  = (const float*)d_in[4];
  const float* bq  = (const float*)d_in[5];
  const float* Wkv = (const float*)d_in[6];
  const float* bkv = (const float*)d_in[7];
  const float* Wo  = (const float*)d_in[8];
  const float* bo  = (const float*)d_in[9];
  const float* Wp  = (const float*)d_in[10];
  const float* bp  = (const float*)d_in[11];
  float* out = (float*)d_out;

  float* ws  = (float*)d_ws;
  float* q   = ws;                                  // [B,H,N,DH]   16 MB
  float* k   = q + (size_t)BB * HH * NN * DH;       // [B,H,S,DH]   24 MB
  float* v   = k + (size_t)BB * HH * SS * DH;       // [B,H,S,DH]   24 MB
  float* pos = v + (size_t)BB * HH * SS * DH;       // [N,DH]       0.5 MB
  float* o   = pos + (size_t)NN * DH;               // [B,H,N,DH]   16 MB

  qproj_k <<<4096,  128, 0, stream>>>(x, Wq, bq, q);
  kvproj_k<<<12288, 128, 0, stream>>>(x, mem, Wkv, bkv, k, v);
  pproj_k <<<128,   128, 0, stream>>>(pe, Wp, bp, pos);
  attn_k  <<<512,   256, 0, stream>>>(q, k, v, pos, em, o);
  oproj_k <<<4096,  128, 0, stream>>>(o, Wo, bo, out);
}